// CompetitiveLayer_40759239639372
// MI455X (gfx1250) — compile-verified
//
#include <hip/hip_runtime.h>

typedef float v2f __attribute__((ext_vector_type(2)));
typedef float v4f __attribute__((ext_vector_type(4)));
typedef float v8f __attribute__((ext_vector_type(8)));

#define NA 8192
#define NB 4096
#define NITERS 50
#define NBLK 128
#define NTHREADS 256
#define NBARRIERS (2 * NITERS)

// Device-wide barrier: one dedicated counter per barrier instance (no reset races).
// All NBLK blocks must be co-resident; 128 blocks x 256 threads with tiny LDS/VGPR
// use is far below WGP residency limits on a flagship CDNA5 part.
__device__ __forceinline__ void grid_barrier(unsigned* ctr) {
    __syncthreads();
    if (threadIdx.x == 0) {
        __threadfence();  // agent-scope release of this block's prior stores
        __hip_atomic_fetch_add(ctr, 1u, __ATOMIC_ACQ_REL, __HIP_MEMORY_SCOPE_AGENT);
        while (__hip_atomic_load(ctr, __ATOMIC_ACQUIRE, __HIP_MEMORY_SCOPE_AGENT) < (unsigned)NBLK) {
            __builtin_amdgcn_s_sleep(2);
        }
        __threadfence();  // agent-scope acquire: invalidate WGP-local cache
    }
    __syncthreads();
}

__global__ void comp_init(const float* __restrict__ BT, float* BF, unsigned* ctrs) {
    const int idx = blockIdx.x * blockDim.x + threadIdx.x;
    if (idx < NB) BF[idx] = BT[idx];
    if (idx < NBARRIERS) ctrs[idx] = 0u;
}

__global__ __launch_bounds__(NTHREADS) void comp_solver(
    const float* __restrict__ AT, const float* __restrict__ BT,
    const float* __restrict__ K, float* __restrict__ C,
    float* AF, float* BF, unsigned* ctrs)
{
    __shared__ float red[8][16];

    const int tid   = threadIdx.x;
    const int wave  = tid >> 5;
    const int lane  = tid & 31;
    const int lhalf = lane >> 4;   // 0: lanes 0-15, 1: lanes 16-31
    const int lrow  = lane & 15;
    // Branch-free operand mask: only lanes 0 and 16 carry the vector operand.
    const float msk = (lrow == 0) ? 1.0f : 0.0f;

    int bar = 0;
    for (int it = 0; it < NITERS; ++it) {
        // ---------- AF phase: AF = AT / (1 + K @ BF) ----------
        // 512 row-groups of 16 rows; 2 waves per group split k (2048 each).
        {
            const int g    = blockIdx.x * 4 + (wave >> 1);
            const int half = wave & 1;
            const int row0 = g << 4;
            const int kbeg = half * (NB / 2);
            // A tile: 16x4 block of K in native WMMA A layout (b64/lane).
            const float* kp  = K + (long)(row0 + lrow) * NB + kbeg + lhalf * 2;
            // B: BF segment, broadcast-loaded by all lanes, masked to column 0.
            const float* bfp = BF + kbeg + lhalf * 2;
            v8f c = {0.f, 0.f, 0.f, 0.f, 0.f, 0.f, 0.f, 0.f};
            #pragma unroll 8
            for (int s = 0; s < (NB / 2) / 4; ++s) {
                v2f a = *(const v2f*)kp;
                v2f t = *(const v2f*)bfp;
                v2f b;
                b.x = t.x * msk;
                b.y = t.y * msk;
                c = __builtin_amdgcn_wmma_f32_16x16x4_f32(
                        false, a, false, b, (short)0, c, false, false);
                kp  += 4;
                bfp += 4;
            }
            // D column 0: lane 0 holds rows M=0..7 (c[0..7]), lane 16 holds M=8..15.
            if (lrow == 0) {
                #pragma unroll
                for (int r = 0; r < 8; ++r) red[wave][lhalf * 8 + r] = c[r];
            }
            __syncthreads();
            if (tid < 64) {
                const int g2 = tid >> 4, r = tid & 15;
                const float s = red[g2 * 2][r] + red[g2 * 2 + 1][r];
                const int row = (blockIdx.x * 4 + g2) * 16 + r;
                AF[row] = AT[row] / (1.0f + s);
            }
            grid_barrier(&ctrs[bar++]);
        }
        // ---------- BF phase: BF = BT / (1 + K^T @ AF) ----------
        // 256 col-groups of 16 cols; 4 waves per group split rows (2048 each).
        {
            const int g    = blockIdx.x * 2 + (wave >> 2);
            const int q    = wave & 3;
            const int col  = (g << 4) + lrow;
            const int kbeg = q * (NA / 4);
            // A: AF segment, broadcast-loaded, masked to row 0.
            const float* afp = AF + kbeg + lhalf * 2;
            // B tile: 4x16 block of K, natural row-major (contiguous per 16-lane group).
            const float* kp0 = K + (long)(kbeg + lhalf * 2) * NB + col;
            v8f c = {0.f, 0.f, 0.f, 0.f, 0.f, 0.f, 0.f, 0.f};
            #pragma unroll 8
            for (int s = 0; s < (NA / 4) / 4; ++s) {
                v2f t = *(const v2f*)afp;
                v2f a;
                a.x = t.x * msk;
                a.y = t.y * msk;
                v2f b;
                b.x = kp0[0];
                b.y = kp0[NB];
                c = __builtin_amdgcn_wmma_f32_16x16x4_f32(
                        false, a, false, b, (short)0, c, false, false);
                afp += 4;
                kp0 += (long)4 * NB;
            }
            // D row 0: VGPR0, lanes 0-15 (N = lane) holds the 16 column partials.
            if (lane < 16) red[wave][lane] = c[0];
            __syncthreads();
            if (tid < 32) {
                const int g2 = tid >> 4, n = tid & 15;
                const float s = red[g2 * 4][n] + red[g2 * 4 + 1][n]
                              + red[g2 * 4 + 2][n] + red[g2 * 4 + 3][n];
                const int cidx = (blockIdx.x * 2 + g2) * 16 + n;
                BF[cidx] = BT[cidx] / (1.0f + s);
            }
            grid_barrier(&ctrs[bar++]);
        }
    }
    // ---------- C = AF_i * K_ij * BF_j (float4-vectorized streaming write) ----------
    {
        const long total4 = (long)NA * (NB / 4);
        const long stride = (long)NBLK * NTHREADS;
        for (long idx = (long)blockIdx.x * NTHREADS + tid; idx < total4; idx += stride) {
            const long j4 = idx % (NB / 4);
            const long i  = idx / (NB / 4);
            const v4f k4 = *(const v4f*)(K + i * NB + j4 * 4);
            const v4f b4 = *(const v4f*)(BF + j4 * 4);
            const float af = AF[i];
            v4f out = k4 * b4 * af;
            *(v4f*)(C + i * NB + j4 * 4) = out;
        }
    }
}

extern "C" void kernel_launch(void* const* d_in, const int* in_sizes, int n_in,
                              void* d_out, int out_size, void* d_ws, size_t ws_size,
                              hipStream_t stream) {
    (void)in_sizes; (void)n_in; (void)out_size; (void)ws_size;
    const float* AT = (const float*)d_in[0];   // [8192]
    const float* BT = (const float*)d_in[1];   // [4096]
    const float* K  = (const float*)d_in[2];   // [8192 x 4096]
    float* C  = (float*)d_out;                 // [8192 x 4096]
    float* AF = (float*)d_ws;                  // [8192]
    float* BF = AF + NA;                       // [4096]
    unsigned* ctrs = (unsigned*)(BF + NB);     // [100]

    comp_init<<<(NB + 255) / 256, 256, 0, stream>>>(BT, BF, ctrs);
    comp_solver<<<NBLK, NTHREADS, 0, stream>>>(AT, BT, K, C, AF, BF, ctrs);
}